// EdgeConvolution_72301479461281
// MI455X (gfx1250) — compile-verified
//
#include <hip/hip_runtime.h>

typedef __attribute__((ext_vector_type(2))) float v2f;
typedef __attribute__((ext_vector_type(8))) float v8f;
typedef __attribute__((ext_vector_type(4))) float f4v;

#define BB 64       // batches (M)
#define DD 128      // feature dim (N of wmma)
#define KK 128      // output dim
#define NN 50000    // nodes
#define EE 200000   // edges (K of wmma)
#define CHUNK 64    // edges per LDS tile (EE % CHUNK == 0)
#define FSTRIDE 136 // padded LDS stride for feats rows (dwords)
#define BSTRIDE 68  // padded LDS stride for bei rows (dwords)
#define MAIN_BLOCKS 1024

#if __has_builtin(__builtin_amdgcn_global_load_async_to_lds_b128) && \
    __has_builtin(__builtin_amdgcn_s_wait_asynccnt)
#define HAS_ASYNC_LDS 1
#else
#define HAS_ASYNC_LDS 0
#endif

#if HAS_ASYNC_LDS
typedef int v4i_gcc __attribute__((vector_size(16)));
typedef __attribute__((address_space(1))) v4i_gcc* gv4i_ptr;  // global (prints as __device__)
typedef __attribute__((address_space(3))) v4i_gcc* lv4i_ptr;  // LDS
__device__ __forceinline__ void async_copy_b128(const float* gsrc, float* lds_dst) {
  __builtin_amdgcn_global_load_async_to_lds_b128(
      (gv4i_ptr)(void*)gsrc, (lv4i_ptr)lds_dst, 0, 0);
}
#endif

// ---------------- prep: zero accumulators (+ optional edge_nodes transpose) --------
__global__ __launch_bounds__(256) void edgeconv_prep(
    const float* __restrict__ edge_nodes,  // [B,N]
    float* __restrict__ ent,               // [N,B]
    float* __restrict__ g_raw,             // [B,D]
    float* __restrict__ row_sum,           // [B]
    int use_ent)
{
  size_t idx = (size_t)blockIdx.x * blockDim.x + threadIdx.x;
  if (idx < (size_t)BB * DD) g_raw[idx] = 0.0f;
  if (idx < BB) row_sum[idx] = 0.0f;
  if (use_ent) {
    const size_t total = (size_t)NN * BB;
    const size_t step = (size_t)gridDim.x * blockDim.x;
    for (size_t i = idx; i < total; i += step) {
      int n = (int)(i >> 6);
      int b = (int)(i & 63);
      ent[i] = edge_nodes[(size_t)b * NN + n];
    }
  }
}

// ---------------- main: g_raw = bei_raw @ edge_feats via V_WMMA_F32_16X16X4_F32 ----
// Feats are streamed chunk-by-chunk into double-buffered LDS with
// GLOBAL_LOAD_ASYNC_TO_LDS_B128 (ASYNCcnt-tracked), overlapping the WMMA work
// on the previous chunk.
__global__ __launch_bounds__(256) void edgeconv_main(
    const float* __restrict__ edge_nodes,  // [B,N]
    const float* __restrict__ edge_feats,  // [E,D]
    const int*   __restrict__ edge_u,      // [E]
    const int*   __restrict__ edge_v,      // [E]
    const float* __restrict__ ent,         // [N,B] (optional)
    float* __restrict__ g_raw,             // [B,D] accumulated via atomics
    float* __restrict__ row_sum,           // [B]   accumulated via atomics
    int use_ent)
{
  __shared__ float sF[2][CHUNK * FSTRIDE]; // double-buffered feats tiles
  __shared__ float sB[BB * BSTRIDE];       // bei tile [64 b][64 edges]

  const int tid  = threadIdx.x;
  const int wave = tid >> 5;               // 0..7 -> owns d range [16*wave, 16*wave+16)
  const int lane = tid & 31;
  const int l16  = lane & 15;
  const int khalf = (lane >> 4) << 1;      // 0 (lanes 0-15) or 2 (lanes 16-31)

  // copy/bei mapping: 4 threads per edge-row / batch-row
  const int row = tid >> 2;                // 0..63
  const int q   = tid & 3;                 // 0..3

  v8f C0 = {}, C1 = {}, C2 = {}, C3 = {};  // 4 M-tiles of 16x16 f32 accumulators
  float local_rs = 0.0f;

  const int nchunks = EE / CHUNK;          // 3125
  int p = 0;                               // LDS buffer parity

#if HAS_ASYNC_LDS
  // prologue: kick off the first chunk's feats into buffer 0 (8 b128 per thread)
  if ((int)blockIdx.x < nchunks) {
    const float* src = edge_feats + (size_t)(blockIdx.x * CHUNK + row) * DD + q * 32;
    float* dst = &sF[0][row * FSTRIDE + q * 32];
#pragma unroll
    for (int j = 0; j < 8; ++j) async_copy_b128(src + 4 * j, dst + 4 * j);
  }
#endif

  for (int c = blockIdx.x; c < nchunks; c += gridDim.x) {
    const int e0 = c * CHUNK;

    // ---- bei_raw tile: gathers of the two endpoint indicator columns ----
    {
#pragma unroll 4
      for (int k = 0; k < 16; ++k) {
        int e = e0 + q * 16 + k;
        int u = edge_u[e];
        int v = edge_v[e];
        float val;
        if (use_ent)
          val = ent[(size_t)u * BB + row] + ent[(size_t)v * BB + row];
        else
          val = edge_nodes[(size_t)row * NN + u] + edge_nodes[(size_t)row * NN + v];
        sB[row * BSTRIDE + q * 16 + k] = val;
        local_rs += val;
      }
    }

    const int cn = c + gridDim.x;
#if HAS_ASYNC_LDS
    // ---- issue next chunk into the other buffer, then wait for this chunk ----
    if (cn < nchunks) {
      const float* src = edge_feats + (size_t)(cn * CHUNK + row) * DD + q * 32;
      float* dst = &sF[1 - p][row * FSTRIDE + q * 32];
#pragma unroll
      for (int j = 0; j < 8; ++j) async_copy_b128(src + 4 * j, dst + 4 * j);
      __builtin_amdgcn_s_wait_asynccnt(8);   // buffer p's 8 loads done (in-order)
    } else {
      __builtin_amdgcn_s_wait_asynccnt(0);
    }
#else
    // ---- synchronous staging fallback (float4 via VGPRs) ----
    {
      const float* src = edge_feats + (size_t)(e0 + row) * DD + q * 32;
      float* dst = &sF[p][row * FSTRIDE + q * 32];
#pragma unroll
      for (int j = 0; j < 8; ++j) {
        f4v val = *(const f4v*)(src + 4 * j);
        *(f4v*)(dst + 4 * j) = val;
      }
      if (cn < nchunks)
        __builtin_prefetch(edge_feats + (size_t)(cn * CHUNK + row) * DD + q * 32, 0, 1);
    }
#endif
    __syncthreads();

    // ---- WMMA: 16 groups of K=4 edges; wave covers N-tile=wave, all 4 M-tiles ----
    const float* fbuf = sF[p];
    const int dcol = wave * 16 + l16;
#pragma unroll 4
    for (int g = 0; g < 16; ++g) {
      const int ek = g * 4 + khalf;
      v2f b;                               // B 4x16: VGPR0 = K{0|2}, VGPR1 = K{1|3}
      b.x = fbuf[ek * FSTRIDE + dcol];
      b.y = fbuf[(ek + 1) * FSTRIDE + dcol];
      v2f a0, a1, a2, a3;                  // A 16x4: lane = M, VGPR0/1 = K pair
      a0.x = sB[(     l16) * BSTRIDE + ek]; a0.y = sB[(     l16) * BSTRIDE + ek + 1];
      a1.x = sB[(16 + l16) * BSTRIDE + ek]; a1.y = sB[(16 + l16) * BSTRIDE + ek + 1];
      a2.x = sB[(32 + l16) * BSTRIDE + ek]; a2.y = sB[(32 + l16) * BSTRIDE + ek + 1];
      a3.x = sB[(48 + l16) * BSTRIDE + ek]; a3.y = sB[(48 + l16) * BSTRIDE + ek + 1];
      C0 = __builtin_amdgcn_wmma_f32_16x16x4_f32(false, a0, false, b, (short)0, C0, false, false);
      C1 = __builtin_amdgcn_wmma_f32_16x16x4_f32(false, a1, false, b, (short)0, C1, false, false);
      C2 = __builtin_amdgcn_wmma_f32_16x16x4_f32(false, a2, false, b, (short)0, C2, false, false);
      C3 = __builtin_amdgcn_wmma_f32_16x16x4_f32(false, a3, false, b, (short)0, C3, false, false);
    }
    __syncthreads();
    p ^= 1;
  }

  // ---- flush partials ----
  atomicAdd(&row_sum[row], local_rs);
  const int dcol  = wave * 16 + l16;
  const int mbase = (lane >> 4) << 3;      // C layout: VGPR j holds rows j and j+8
#pragma unroll
  for (int j = 0; j < 8; ++j) {
    atomicAdd(&g_raw[(0 * 16 + mbase + j) * DD + dcol], C0[j]);
    atomicAdd(&g_raw[(1 * 16 + mbase + j) * DD + dcol], C1[j]);
    atomicAdd(&g_raw[(2 * 16 + mbase + j) * DD + dcol], C2[j]);
    atomicAdd(&g_raw[(3 * 16 + mbase + j) * DD + dcol], C3[j]);
  }
}

// ---------------- finalize: h = inv[b] * (g_raw @ W) + s[b]*bias --------------------
__global__ __launch_bounds__(128) void edgeconv_final(
    const float* __restrict__ g_raw,   // [B,D]
    const float* __restrict__ weight,  // [D,K] row-major
    const float* __restrict__ bias,    // [K]
    const float* __restrict__ row_sum, // [B]
    float* __restrict__ out)           // [B,K]
{
  const int b = blockIdx.x;            // 0..63
  const int k = threadIdx.x;           // 0..127
  __shared__ float sg[DD];
  sg[k] = g_raw[b * DD + k];
  __syncthreads();
  const float rs  = row_sum[b];
  const float inv = rs > 0.0f ? 1.0f / rs : 0.0f;
  const float s   = rs > 0.0f ? 1.0f : 0.0f;
  float acc = 0.0f;
#pragma unroll 8
  for (int d = 0; d < DD; ++d) acc += sg[d] * weight[d * KK + k];
  out[b * KK + k] = inv * acc + s * bias[k];
}

extern "C" void kernel_launch(void* const* d_in, const int* in_sizes, int n_in,
                              void* d_out, int out_size, void* d_ws, size_t ws_size,
                              hipStream_t stream) {
  const float* edge_nodes = (const float*)d_in[0];  // [B,N]
  const float* edge_feats = (const float*)d_in[1];  // [E,D]
  const float* weight     = (const float*)d_in[2];  // [D,K]
  const float* bias       = (const float*)d_in[3];  // [K]
  const int*   edge_u     = (const int*)d_in[4];    // [E]
  const int*   edge_v     = (const int*)d_in[5];    // [E]
  float* out = (float*)d_out;                       // [B,K]

  // workspace layout: row_sum (64 f) | g_raw (64*128 f) | ent (N*64 f, optional)
  char* ws = (char*)d_ws;
  float* row_sum = (float*)ws;
  float* g_raw   = (float*)(ws + 256);
  float* ent     = (float*)(ws + 256 + (size_t)BB * DD * sizeof(float));
  const size_t need_ent = 256 + (size_t)BB * DD * sizeof(float)
                        + (size_t)NN * BB * sizeof(float);
  const int use_ent = (ws_size >= need_ent) ? 1 : 0;

  const int prep_blocks = use_ent ? 2048 : 32;
  edgeconv_prep<<<prep_blocks, 256, 0, stream>>>(edge_nodes, ent, g_raw, row_sum, use_ent);
  edgeconv_main<<<MAIN_BLOCKS, 256, 0, stream>>>(edge_nodes, edge_feats, edge_u, edge_v,
                                                 ent, g_raw, row_sum, use_ent);
  edgeconv_final<<<BB, KK, 0, stream>>>(g_raw, weight, bias, row_sum, out);
}